// NonGridProjectLayer_17463337026052
// MI455X (gfx1250) — compile-verified
//
#include <hip/hip_runtime.h>

// Problem constants (match reference)
#define NB 4
#define NV 5
#define NC 128
#define NH 128
#define NW 128
#define NP 4096

typedef __attribute__((ext_vector_type(2))) float v2f;
typedef __attribute__((ext_vector_type(8))) float v8f;

// ---------------------------------------------------------------------------
// Kernel 0: zero the per-point NaN flags (ws is poisoned; must reinit per call)
// ---------------------------------------------------------------------------
__global__ void init_flags_kernel(int* __restrict__ nanflag) {
    int t = blockIdx.x * blockDim.x + threadIdx.x;
    if (t < NB * NP) nanflag[t] = 0;
}

// ---------------------------------------------------------------------------
// Kernel 1: projection. Each wave projects 16 points of one (n,v) using the
// fp32 WMMA 16x16x4:  D[m][i] = sum_k A[m][k] * B[k][i],
//   A[m][k] = (pt[m][k] - T[k]) for k<3, 0 for k=3
//   B[k][i] = R[i][k]           (i<3; zero elsewhere)
// Exact fp32 math; shared-B GEMM is legitimate since R,T are per-(n,v).
// Block = 256 threads = 8 waves = 128 points.
// ---------------------------------------------------------------------------
__global__ void proj_kernel(const float* __restrict__ sp,     // (N,P,5)
                            const float* __restrict__ R,      // (N,V,3,3)
                            const float* __restrict__ T,      // (N,V,3)
                            const float* __restrict__ f,      // (N,V,2)
                            const float* __restrict__ c,      // (N,V,2)
                            const float* __restrict__ kk,     // (N,V,3)
                            const float* __restrict__ pp,     // (N,V,2)
                            const float* __restrict__ trans,  // (N,V,2,3)
                            const float* __restrict__ wh,     // (N,V,2)
                            const float* __restrict__ fm_size,// (2,)
                            float* __restrict__ gx,           // (N*V*P)
                            float* __restrict__ gy,           // (N*V*P)
                            float* __restrict__ bnd)          // (N*V*P)
{
    const int tilesPerNV = NP / 128;            // 32
    int b     = blockIdx.x;
    int nv    = b / tilesPerNV;
    int ptile = b % tilesPerNV;
    int n = nv / NV;

    int tid  = threadIdx.x;
    int wave = tid >> 5;
    int lane = tid & 31;
    int m    = lane & 15;     // point within wave-tile (A row / B-D col index)
    int kh   = lane >> 4;     // K-half (0: K=0,1 ; 1: K=2,3)
    int p    = ptile * 128 + wave * 16 + m;

    __shared__ float xcs[8][16][3];

    // ---- A operand: 16x4 f32, 2 VGPRs/lane ----
    const float* Tv  = T + nv * 3;
    const float* spp = sp + ((size_t)n * NP + p) * 5;
    v2f a;
    if (kh == 0) { a.x = spp[0] - Tv[0]; a.y = spp[1] - Tv[1]; }
    else         { a.x = spp[2] - Tv[2]; a.y = 0.0f;           }

    // ---- B operand: 4x16 f32, 2 VGPRs/lane; B[k][i] = R[i][k] ----
    const float* Rv = R + nv * 9;
    v2f bb; bb.x = 0.0f; bb.y = 0.0f;
    if (m < 3) {
        if (kh == 0) { bb.x = Rv[m * 3 + 0]; bb.y = Rv[m * 3 + 1]; }
        else         { bb.x = Rv[m * 3 + 2]; bb.y = 0.0f;          }
    }

    v8f acc = {};
    acc = __builtin_amdgcn_wmma_f32_16x16x4_f32(
        /*neg_a=*/false, a, /*neg_b=*/false, bb,
        /*c_mod=*/(short)0, acc, /*reuse_a=*/false, /*reuse_b=*/false);

    // D layout: VGPR r, lane L -> row (8*kh + r) = point idx, col (L&15) = Xc comp
    if (m < 3) {
#pragma unroll
        for (int r = 0; r < 8; ++r) xcs[wave][kh * 8 + r][m] = acc[r];
    }
    __syncthreads();

    if (lane < 16) {
        float X = xcs[wave][m][0];
        float Y = xcs[wave][m][1];
        float Z = xcs[wave][m][2];
        float xn = X / Z;
        float yn = Y / Z;
        float r2 = xn * xn + yn * yn;

        const float* kv = kk + nv * 3;
        const float* pv = pp + nv * 2;
        float radial = 1.0f + kv[0] * r2 + kv[1] * r2 * r2 + kv[2] * r2 * r2 * r2;
        float xd = xn * radial + 2.0f * pv[0] * xn * yn + pv[1] * (r2 + 2.0f * xn * xn);
        float yd = yn * radial + pv[0] * (r2 + 2.0f * yn * yn) + 2.0f * pv[1] * xn * yn;

        const float* fv = f + nv * 2;
        const float* cv = c + nv * 2;
        float xy0 = xd * fv[0] + cv[0];
        float xy1 = yd * fv[1] + cv[1];

        const float* whv = wh + nv * 2;
        float width  = whv[0];
        float height = whv[1];
        float bound  = (xy0 >= 0.0f && xy1 >= 0.0f && xy0 < width && xy1 < height) ? 1.0f : 0.0f;

        float mx = fmaxf(width, height);
        float px = fminf(fmaxf(xy0, -1.0f), mx);
        float py = fminf(fmaxf(xy1, -1.0f), mx);

        const float* tv = trans + nv * 6;
        float pt0 = tv[0] * px + tv[1] * py + tv[2];
        float pt1 = tv[3] * px + tv[4] * py + tv[5];

        float fmw = fm_size[0], fmh = fm_size[1];
        float nx = fminf(fmaxf(pt0 / (fmw - 1.0f) * 2.0f - 1.0f, -1.1f), 1.1f);
        float ny = fminf(fmaxf(pt1 / (fmh - 1.0f) * 2.0f - 1.0f, -1.1f), 1.1f);

        // grid_sample pixel space of the (H,W) feature map
        float gxv = (nx + 1.0f) * 0.5f * (float)(NW - 1);
        float gyv = (ny + 1.0f) * 0.5f * (float)(NH - 1);

        int idx = nv * NP + p;
        gx[idx]  = gxv;
        gy[idx]  = gyv;
        bnd[idx] = bound;
    }
}

// ---------------------------------------------------------------------------
// Kernel 2: bilinear sampling. One 128-thread block per (n,p,v); thread = ch.
// ---------------------------------------------------------------------------
__device__ __forceinline__ float fetch_px(const float* __restrict__ img, int xi, int yi) {
    bool valid = (xi >= 0) & (xi <= NW - 1) & (yi >= 0) & (yi <= NH - 1);
    int xc = min(max(xi, 0), NW - 1);
    int yc = min(max(yi, 0), NH - 1);
    float v = img[yc * NW + xc];
    return valid ? v : 0.0f;
}

__global__ void sample_kernel(const float* __restrict__ fm,   // (N,V,C,H,W)
                              const float* __restrict__ gx,
                              const float* __restrict__ gy,
                              const float* __restrict__ bnd,
                              float* __restrict__ feats,      // (N,P,V,C)
                              int* __restrict__ nanflag)      // (N*P)
{
    int b  = blockIdx.x;               // ((n*P + p)*V + v)
    int v  = b % NV;
    int np = b / NV;
    int p  = np % NP;
    int n  = np / NP;
    int ch = threadIdx.x;

    int widx    = (n * NV + v) * NP + p;
    float x     = gx[widx];
    float y     = gy[widx];
    float bound = bnd[widx];

    float x0f = floorf(x), y0f = floorf(y);
    int   x0  = (int)x0f,  y0  = (int)y0f;
    float wx1 = x - x0f,   wy1 = y - y0f;
    float wx0 = 1.0f - wx1, wy0 = 1.0f - wy1;

    const float* img = fm + ((size_t)((n * NV + v) * NC + ch)) * (NH * NW);

    // Prefetch the second image row while the first row's loads are in flight.
    {
        int yc1 = min(max(y0 + 1, 0), NH - 1);
        int xc0 = min(max(x0, 0), NW - 1);
        __builtin_prefetch(img + yc1 * NW + xc0, 0, 0);
    }

    float f00 = fetch_px(img, x0,     y0);
    float f10 = fetch_px(img, x0 + 1, y0);
    float f01 = fetch_px(img, x0,     y0 + 1);
    float f11 = fetch_px(img, x0 + 1, y0 + 1);

    float s = f00 * (wx0 * wy0) + f10 * (wx1 * wy0)
            + f01 * (wx0 * wy1) + f11 * (wx1 * wy1);
    float val = s * bound;   // NaN propagates like the reference (0*NaN = NaN)

    if (__builtin_isnan(val)) atomicOr(&nanflag[n * NP + p], 1);

    feats[((size_t)(n * NP + p) * NV + v) * NC + ch] = val;
}

// ---------------------------------------------------------------------------
// Kernel 3: bounding = (sum_v bound > 0) && no NaN in feats[n,p,:,:]
// ---------------------------------------------------------------------------
__global__ void bounding_kernel(const float* __restrict__ bnd,
                                const int* __restrict__ nanflag,
                                float* __restrict__ bounding) {
    int t = blockIdx.x * blockDim.x + threadIdx.x;
    if (t >= NB * NP) return;
    int n = t / NP;
    int p = t % NP;
    float s = 0.0f;
#pragma unroll
    for (int v = 0; v < NV; ++v) s += bnd[(n * NV + v) * NP + p];
    bounding[t] = (s > 0.0f && nanflag[t] == 0) ? 1.0f : 0.0f;
}

// ---------------------------------------------------------------------------
extern "C" void kernel_launch(void* const* d_in, const int* in_sizes, int n_in,
                              void* d_out, int out_size, void* d_ws, size_t ws_size,
                              hipStream_t stream) {
    (void)in_sizes; (void)n_in; (void)out_size; (void)ws_size;

    const float* fm      = (const float*)d_in[0];
    const float* sp      = (const float*)d_in[1];
    const float* R       = (const float*)d_in[2];
    const float* T       = (const float*)d_in[3];
    const float* f       = (const float*)d_in[4];
    const float* c       = (const float*)d_in[5];
    const float* kk      = (const float*)d_in[6];
    const float* pp      = (const float*)d_in[7];
    const float* trans   = (const float*)d_in[8];
    const float* wh      = (const float*)d_in[9];
    const float* fm_size = (const float*)d_in[10];

    const size_t NVP = (size_t)NB * NV * NP;          // 81920
    float* gx  = (float*)d_ws;
    float* gy  = gx + NVP;
    float* bnd = gy + NVP;
    int*   nanflag = (int*)(bnd + NVP);

    float* feats    = (float*)d_out;                              // N*P*V*C
    float* bounding = feats + (size_t)NB * NP * NV * NC;          // N*P
    float* sp_out   = bounding + (size_t)NB * NP;                 // N*P*5

    // 0) reset NaN flags
    init_flags_kernel<<<(NB * NP + 255) / 256, 256, 0, stream>>>(nanflag);

    // 1) projection (WMMA f32 16x16x4): 8 waves x 16 points per block
    {
        dim3 grid(NB * NV * (NP / 128));
        proj_kernel<<<grid, 256, 0, stream>>>(sp, R, T, f, c, kk, pp, trans, wh,
                                              fm_size, gx, gy, bnd);
    }

    // 2) bilinear sampling: one block per (n,p,v), thread = channel
    {
        dim3 grid((unsigned)((size_t)NB * NP * NV));
        sample_kernel<<<grid, NC, 0, stream>>>(fm, gx, gy, bnd, feats, nanflag);
    }

    // 3) bounding reduction
    bounding_kernel<<<(NB * NP + 255) / 256, 256, 0, stream>>>(bnd, nanflag, bounding);

    // 4) sample_points passthrough
    hipMemcpyAsync(sp_out, sp, (size_t)NB * NP * 5 * sizeof(float),
                   hipMemcpyDeviceToDevice, stream);
}